// SEIR_model_24902220382875
// MI455X (gfx1250) — compile-verified
//
#include <hip/hip_runtime.h>
#include <hip/hip_bf16.h>

// SEIR sequential scan, MI455X (gfx1250, wave32).
// Single wave runs the serial nonlinear recurrence; Tensor Data Mover (TDM)
// streams the 4 parameter arrays into double-buffered LDS one chunk ahead.
// Broadcasts of per-step coefficients use v_readlane (SGPR result) so the
// dependent chain is only 3 FMAs/step and no DS traffic is on the hot loop.

#define DATE_LEN 1048576
#define SEIR_STEPS (DATE_LEN - 1)      // 1048575
#define CH 1024                        // params per chunk per array
#define NCHUNKS ((SEIR_STEPS + CH - 1) / CH)  // 1024

typedef unsigned int u32x4 __attribute__((ext_vector_type(4)));
typedef int          i32x4 __attribute__((ext_vector_type(4)));
typedef int          i32x8 __attribute__((ext_vector_type(8)));

// Broadcast lane `i`'s value to all lanes via v_readlane_b32 (result lands in
// an SGPR and feeds VALU ops as a scalar operand -- no LDS, no DScnt waits).
__device__ __forceinline__ float bcast(float x, int i) {
  return __int_as_float(__builtin_amdgcn_readlane(__float_as_int(x), i));
}

// Issue one TDM load of a 1-D row of `nelem` f32 from global into LDS.
// D# layout per CDNA5 ISA ch.8: group0 = {count/lds_addr/global_addr/type},
// group1 = {data_size, tensor_dim0/1, tile_dim0, stride0}.
__device__ __forceinline__ void tdm_load_row(const float* gsrc,
                                             unsigned lds_off,
                                             unsigned nelem) {
  unsigned long long ga = (unsigned long long)(uintptr_t)gsrc;
  u32x4 g0;
  g0[0] = 1u;                                       // count=1 (valid), user mode
  g0[1] = lds_off;                                  // LDS byte address
  g0[2] = (unsigned)(ga & 0xFFFFFFFFu);             // global_addr[31:0]
  g0[3] = ((unsigned)(ga >> 32) & 0x01FFFFFFu)      // global_addr[56:32]
          | 0x80000000u;                            // type = 2 ("image")
  i32x8 g1;
  g1[0] = 0x00020000;                               // data_size=2 -> 4 bytes
  g1[1] = (int)((nelem & 0xFFFFu) << 16);           // tensor_dim0[15:0]
  g1[2] = (int)(((nelem >> 16) & 0xFFFFu)           // tensor_dim0[31:16]
          | (1u << 16));                            // tensor_dim1 = 1
  g1[3] = (int)((nelem & 0xFFFFu) << 16);           // tile_dim0 = nelem
  g1[4] = 0;                                        // tile_dim1/2 unused
  g1[5] = (int)nelem;                               // tensor_dim0_stride lo
  g1[6] = 0;
  g1[7] = 0;
  i32x4 z4 = {0, 0, 0, 0};
#if defined(__clang_major__) && (__clang_major__ >= 23)
  i32x8 z8 = {0, 0, 0, 0, 0, 0, 0, 0};
  __builtin_amdgcn_tensor_load_to_lds(g0, g1, z4, z4, z8, 0);
#else
  __builtin_amdgcn_tensor_load_to_lds(g0, g1, z4, z4, 0);
#endif
}

__global__ __launch_bounds__(32)
void seir_scan_kernel(const float* __restrict__ beta,
                      const float* __restrict__ gamma2,
                      const float* __restrict__ theta,
                      const float* __restrict__ alpha,
                      float* __restrict__ out) {
  __shared__ float sb[2][4][CH];   // double buffer x {beta,gamma,theta,alpha}
  const int lane = (int)threadIdx.x;   // block = exactly one wave32
  const float invN = 1.0f / 2870000.0f;

  // State (replicated in all lanes). S0 = N - I0 - E0 - R0 - D0.
  float S = 2869834.0f, I = 41.0f, E = 123.0f, R = 2.0f, D = 0.0f;

  if (lane == 0) {
    // Row 0 of each trajectory = initial state.
    out[0 * DATE_LEN] = S;
    out[1 * DATE_LEN] = I;
    out[2 * DATE_LEN] = E;
    out[3 * DATE_LEN] = R;
    out[4 * DATE_LEN] = D;
    // Scalars: beta[-1] (post-override = beta[STEPS-2]), gamma_2[-1].
    out[5 * DATE_LEN + 0] = beta[SEIR_STEPS - 2];
    out[5 * DATE_LEN + 1] = gamma2[SEIR_STEPS - 1];
  }

  const float* srcs[4] = {beta, gamma2, theta, alpha};

  // Prime the pipeline: TDM-load chunk 0 into buffer 0.
  {
    unsigned n0 = (SEIR_STEPS < CH) ? (unsigned)SEIR_STEPS : (unsigned)CH;
    #pragma unroll
    for (int p = 0; p < 4; ++p)
      tdm_load_row(srcs[p], (unsigned)(uintptr_t)(&sb[0][p][0]), n0);
  }

  for (unsigned c = 0; c < NCHUNKS; ++c) {
    unsigned base = c * CH;
    unsigned n = (unsigned)SEIR_STEPS - base;
    if (n > CH) n = CH;

    // Prefetch chunk c+1 into the other buffer, then wait for chunk c.
    if (c + 1 < NCHUNKS) {
      unsigned nb = (unsigned)SEIR_STEPS - (base + CH);
      if (nb > CH) nb = CH;
      unsigned bi = (c + 1) & 1u;
      #pragma unroll
      for (int p = 0; p < 4; ++p)
        tdm_load_row(srcs[p] + base + CH,
                     (unsigned)(uintptr_t)(&sb[bi][p][0]), nb);
      __builtin_amdgcn_s_wait_tensorcnt(4);  // 4 newest in flight; chunk c done
    } else {
      __builtin_amdgcn_s_wait_tensorcnt(0);
    }
    asm volatile("" ::: "memory");  // LDS written by TDM; fence the compiler

    const float (*buf)[CH] = sb[c & 1u];

    unsigned g = 0;
    // -------- fast path: full groups of 32 steps --------
    for (; g + 32u <= n; g += 32u) {
      unsigned li = g + (unsigned)lane;
      unsigned gi = base + li;
      // beta[-1] <- beta[-2] override (only ever fires in the global tail,
      // kept here for safety; costs 2 off-chain VALU).
      unsigned lib = (gi == (unsigned)(SEIR_STEPS - 1)) ? (li - 1u) : li;
      float b  = buf[0][lib];
      float gm = buf[1][li];
      float th = buf[2][li];
      float al = buf[3][li];
      // Per-lane (off-chain) coefficient prep.
      float cb = b * b * invN;          // beta^2 / N
      float ca = al * al;               // alpha^2
      float cg = gm * gm;               // gamma^2
      float ct = th * th;               // theta^2
      float ea = 1.0f - ca;             // E' = fma(E, ea, new_inf)
      float kk = 1.0f - cg - ct;        // I' = fma(I, kk, ca*E)
      float oS = S, oI = I, oE = E, oR = R, oD = D;
      #pragma unroll
      for (int i = 0; i < 32; ++i) {
        // v_readlane broadcasts: off the dependency chain, SGPR results.
        float cbi = bcast(cb, i);
        float cai = bcast(ca, i);
        float cgi = bcast(cg, i);
        float cti = bcast(ct, i);
        float eai = bcast(ea, i);
        float kki = bcast(kk, i);
        // Critical path: 3 dependent mul/FMA per step.
        float u  = cbi * S;             // (beta^2/N) * S
        float nl = cai * E;             // alpha^2 * E
        float ni = u * I;               // new infections
        float Rn = fmaf(cgi, I, R);
        float Dn = fmaf(cti, I, D);
        float In = fmaf(I, kki, nl);    // I + new_lat - rec - die
        float Sn = fmaf(-u, I, S);      // S - new_inf
        float En = fmaf(E, eai, ni);    // E + new_inf - new_lat
        S = Sn; I = In; E = En; R = Rn; D = Dn;
        bool me = (lane == i);
        oS = me ? S : oS;
        oI = me ? I : oI;
        oE = me ? E : oE;
        oR = me ? R : oR;
        oD = me ? D : oD;
      }
      unsigned pos = 1u + gi;           // trajectory row for this lane
      out[0 * DATE_LEN + pos] = oS;
      out[1 * DATE_LEN + pos] = oI;
      out[2 * DATE_LEN + pos] = oE;
      out[3 * DATE_LEN + pos] = oR;
      out[4 * DATE_LEN + pos] = oD;
    }

    // -------- tail: m < 32 steps (executes once, at the very end) --------
    unsigned m = n - g;
    if (m) {
      unsigned li = g + (unsigned)lane;
      if (li > n - 1u) li = n - 1u;     // clamp; clamped lanes never broadcast
      unsigned gi = base + li;
      unsigned lib = (gi == (unsigned)(SEIR_STEPS - 1)) ? (li - 1u) : li;
      float b  = buf[0][lib];
      float gm = buf[1][li];
      float th = buf[2][li];
      float al = buf[3][li];
      float cb = b * b * invN;
      float ca = al * al;
      float cg = gm * gm;
      float ct = th * th;
      float ea = 1.0f - ca;
      float kk = 1.0f - cg - ct;
      float oS = S, oI = I, oE = E, oR = R, oD = D;
      for (int i = 0; i < (int)m; ++i) {  // i is wave-uniform: readlane legal
        float cbi = bcast(cb, i);
        float cai = bcast(ca, i);
        float cgi = bcast(cg, i);
        float cti = bcast(ct, i);
        float eai = bcast(ea, i);
        float kki = bcast(kk, i);
        float u  = cbi * S;
        float nl = cai * E;
        float ni = u * I;
        float Rn = fmaf(cgi, I, R);
        float Dn = fmaf(cti, I, D);
        float In = fmaf(I, kki, nl);
        float Sn = fmaf(-u, I, S);
        float En = fmaf(E, eai, ni);
        S = Sn; I = In; E = En; R = Rn; D = Dn;
        bool me = (lane == i);
        oS = me ? S : oS;
        oI = me ? I : oI;
        oE = me ? E : oE;
        oR = me ? R : oR;
        oD = me ? D : oD;
      }
      if ((unsigned)lane < m) {
        unsigned pos = 1u + base + g + (unsigned)lane;
        out[0 * DATE_LEN + pos] = oS;
        out[1 * DATE_LEN + pos] = oI;
        out[2 * DATE_LEN + pos] = oE;
        out[3 * DATE_LEN + pos] = oR;
        out[4 * DATE_LEN + pos] = oD;
      }
    }
  }
}

extern "C" void kernel_launch(void* const* d_in, const int* in_sizes, int n_in,
                              void* d_out, int out_size, void* d_ws, size_t ws_size,
                              hipStream_t stream) {
  (void)in_sizes; (void)n_in; (void)out_size; (void)d_ws; (void)ws_size;
  const float* beta   = (const float*)d_in[0];
  const float* gamma2 = (const float*)d_in[1];
  const float* theta  = (const float*)d_in[2];
  const float* alpha  = (const float*)d_in[3];
  // d_in[4] = X, unused by the forward pass.
  float* out = (float*)d_out;
  // One wave: the recurrence is a single serial dependency chain.
  seir_scan_kernel<<<dim3(1), dim3(32), 0, stream>>>(beta, gamma2, theta,
                                                     alpha, out);
}